// KMeans_80719615361356
// MI455X (gfx1250) — compile-verified
//
#include <hip/hip_runtime.h>

typedef __attribute__((ext_vector_type(2))) float v2f;
typedef __attribute__((ext_vector_type(8))) float v8f;

#define B_   4
#define K_   4
#define D_   20
#define XR   24                 // padded LDS row stride (16B-aligned rows)
#define TILE 256
#define MAXBLOCKS 128
#define PSZ  (K_ * (D_ + 1))    // 84 floats per block partial: num[4][20] + den[4]

// ---------------------------------------------------------------------------
// init: means[b,k,:] = data[b, init_idx[b,k], :]
// ---------------------------------------------------------------------------
__global__ __launch_bounds__(320) void kinit(const float* __restrict__ data,
                                             const int* __restrict__ idx,
                                             float* __restrict__ means, int N) {
  int tid = threadIdx.x;
  if (tid < B_ * K_ * D_) {
    int b = tid / (K_ * D_);
    int r = tid % (K_ * D_);
    int k = r / D_;
    int d = r % D_;
    int p = idx[b * K_ + k];
    means[tid] = data[((size_t)b * N + p) * D_ + d];
  }
}

// ---------------------------------------------------------------------------
// accum: per (batch, block) partial of num[k][d] and den[k] using f32 WMMA.
//   D = A x B + C with M=cluster, N=feature(+ones col), contraction = 4 pts.
//   Double-buffered LDS tiles, one barrier per tile; the steady-state loop
//   runs only over full tiles (no bounds guard, no zero-fill); a ragged tail
//   (never taken for the reference shapes) is handled in a separate epilogue.
// ---------------------------------------------------------------------------
__global__ __launch_bounds__(256) void kaccum(const float* __restrict__ data,
                                              const float* __restrict__ means,
                                              const float* __restrict__ scale,
                                              float* __restrict__ partials,
                                              int N) {
  __shared__ float sM[K_ * D_];
  __shared__ float sX[2][TILE][XR];    // cols 0..19 = x, col 20 = ones (den)
  __shared__ float sA[2][TILE][K_];    // one float4 per row
  __shared__ float sRed[8][K_][D_ + 1];

  const int tid = threadIdx.x;
  const int b = blockIdx.y;
  const int w = tid >> 5;        // wave id 0..7
  const int l = tid & 31;        // lane id
  const float sc = scale[0];

  if (tid < K_ * D_) sM[tid] = means[b * K_ * D_ + tid];

  v8f c0 = {};   // num[k][d0..15]
  v8f c1 = {};   // num[k][d16..19], den[k] at col 4

  const int stride = gridDim.x;
  const float* dB = data + (size_t)b * N * D_;

  const int m = l & 15;      // A row (cluster) / B column (feature)
  const int half = l >> 4;   // which pair of contraction rows

  // softmax + LDS publish of one point held in registers
  auto publish = [&](int buf, const float4* xv, bool ok) {
    float4* row = reinterpret_cast<float4*>(&sX[buf][tid][0]);
#pragma unroll
    for (int j = 0; j < 5; ++j) row[j] = xv[j];
    sX[buf][tid][D_] = ok ? 1.0f : 0.0f;

    float xr[D_];
#pragma unroll
    for (int j = 0; j < 5; ++j) {
      xr[4 * j + 0] = xv[j].x; xr[4 * j + 1] = xv[j].y;
      xr[4 * j + 2] = xv[j].z; xr[4 * j + 3] = xv[j].w;
    }
    float logit[K_];
#pragma unroll
    for (int k = 0; k < K_; ++k) {
      float acc = 0.0f;
#pragma unroll
      for (int d = 0; d < D_; ++d) {
        float df = xr[d] - sM[k * D_ + d];
        acc += df * df;
      }
      logit[k] = -acc * sc;
    }
    float mx = logit[0];
#pragma unroll
    for (int k = 1; k < K_; ++k) mx = fmaxf(mx, logit[k]);
    float e[K_]; float sum = 0.0f;
#pragma unroll
    for (int k = 0; k < K_; ++k) { e[k] = __expf(logit[k] - mx); sum += e[k]; }
    float inv = ok ? (1.0f / sum) : 0.0f;
    float4 av;
    av.x = e[0] * inv; av.y = e[1] * inv; av.z = e[2] * inv; av.w = e[3] * inv;
    *reinterpret_cast<float4*>(&sA[buf][tid][0]) = av;
  };

  // full tile: unconditional clause of 5 x global_load_b128 per thread
  auto stageFull = [&](int buf, int t) {
    const int n = t * TILE + tid;
    const float4* xp = reinterpret_cast<const float4*>(dB + (size_t)n * D_);
    float4 xv[5];
#pragma unroll
    for (int j = 0; j < 5; ++j) xv[j] = xp[j];
    publish(buf, xv, true);
  };

  // ragged tail tile (cold path)
  auto stageRagged = [&](int buf, int t) {
    const int n = t * TILE + tid;
    const bool ok = (n < N);
    float4 z; z.x = z.y = z.z = z.w = 0.0f;
    float4 xv[5];
#pragma unroll
    for (int j = 0; j < 5; ++j) xv[j] = z;
    if (ok) {
      const float4* xp = reinterpret_cast<const float4*>(dB + (size_t)n * D_);
#pragma unroll
      for (int j = 0; j < 5; ++j) xv[j] = xp[j];
    }
    publish(buf, xv, ok);
  };

  // WMMA over buffer `buf`: wave w contracts its 32 points, 4 per step.
  // A 16x4 f32: lane = m + 16*(kk/2), vgpr = kk&1
  // B 4x16 f32: lane = n + 16*(kk/2), vgpr = kk&1
  auto mma = [&](int buf) {
#pragma unroll
    for (int g = 0; g < 8; ++g) {
      int base = w * 32 + g * 4;
      int r0 = base + half * 2;
      int r1 = r0 + 1;
      v2f a, b0, b1;
      a.x = (m < K_) ? sA[buf][r0][m] : 0.0f;
      a.y = (m < K_) ? sA[buf][r1][m] : 0.0f;
      b0.x = sX[buf][r0][m];
      b0.y = sX[buf][r1][m];
      int cg1 = 16 + m;        // d-group 1: d=16..19, col 4 = ones (den)
      b1.x = (cg1 <= D_) ? sX[buf][r0][cg1] : 0.0f;
      b1.y = (cg1 <= D_) ? sX[buf][r1][cg1] : 0.0f;
      c0 = __builtin_amdgcn_wmma_f32_16x16x4_f32(false, a, false, b0,
                                                 (short)0, c0, false, false);
      c1 = __builtin_amdgcn_wmma_f32_16x16x4_f32(false, a, false, b1,
                                                 (short)0, c1, false, false);
    }
  };

  // software-pipelined loop over FULL tiles only; single barrier per tile
  const int numFull = N / TILE;
  int t = blockIdx.x;
  int cur = 0;
  if (t < numFull) stageFull(cur, t);
  __syncthreads();
  while (t < numFull) {
    int tn = t + stride;
    if (tn < numFull) stageFull(1 - cur, tn);   // prefetch next tile
    mma(cur);                                    // consume current tile
    __syncthreads();
    cur = 1 - cur;
    t = tn;
  }

  // ragged tail epilogue (block-uniform; never taken when N % TILE == 0)
  if ((N % TILE) != 0 && (numFull % stride) == (int)blockIdx.x) {
    stageRagged(cur, numFull);
    __syncthreads();
    mma(cur);
  }

  __syncthreads();

  // extract per-wave fragments: C/D rows 0..3 live in VGPRs 0..3, lanes 0..15
  if (l < 16) {
#pragma unroll
    for (int k = 0; k < K_; ++k) sRed[w][k][l] = c0[k];
  }
  if (l < 5) {
#pragma unroll
    for (int k = 0; k < K_; ++k) sRed[w][k][16 + l] = c1[k];
  }
  __syncthreads();

  // fixed-order cross-wave reduction -> per-block partial (deterministic)
  if (tid < PSZ) {
    float acc = 0.0f;
#pragma unroll
    for (int ww = 0; ww < 8; ++ww) acc += (&sRed[ww][0][0])[tid];
    partials[((size_t)b * gridDim.x + blockIdx.x) * PSZ + tid] = acc;
  }
}

// ---------------------------------------------------------------------------
// reduce: sum block partials in fixed order, means = num/den.
// ---------------------------------------------------------------------------
__global__ __launch_bounds__(128) void kreduce(const float* __restrict__ partials,
                                               float* __restrict__ means,
                                               float* __restrict__ outMeans,
                                               int nBlocks) {
  __shared__ float sSum[PSZ];
  int tid = threadIdx.x, b = blockIdx.x;
  if (tid < PSZ) {
    float s = 0.0f;
    for (int blk = 0; blk < nBlocks; ++blk)
      s += partials[((size_t)b * nBlocks + blk) * PSZ + tid];
    sSum[tid] = s;
  }
  __syncthreads();
  if (tid < K_ * D_) {
    int k = tid / D_;
    int d = tid % D_;
    float mval = sSum[k * (D_ + 1) + d] / sSum[k * (D_ + 1) + D_];
    means[b * K_ * D_ + tid] = mval;
    if (outMeans) outMeans[b * K_ * D_ + tid] = mval;
  }
}

// ---------------------------------------------------------------------------
// final: assignments with converged means -> d_out [B,N,K]
// ---------------------------------------------------------------------------
__global__ __launch_bounds__(256) void kfinal(const float* __restrict__ data,
                                              const float* __restrict__ means,
                                              const float* __restrict__ scale,
                                              float* __restrict__ out, int N) {
  __shared__ float sM[K_ * D_];
  int tid = threadIdx.x, b = blockIdx.y;
  if (tid < K_ * D_) sM[tid] = means[b * K_ * D_ + tid];
  __syncthreads();
  int n = blockIdx.x * 256 + tid;
  if (n >= N) return;
  const float sc = scale[0];

  const float4* xp = reinterpret_cast<const float4*>(data + ((size_t)b * N + n) * D_);
  float4 xv[5];
#pragma unroll
  for (int j = 0; j < 5; ++j) xv[j] = xp[j];
  float xr[D_];
#pragma unroll
  for (int j = 0; j < 5; ++j) {
    xr[4 * j + 0] = xv[j].x; xr[4 * j + 1] = xv[j].y;
    xr[4 * j + 2] = xv[j].z; xr[4 * j + 3] = xv[j].w;
  }

  float logit[K_];
#pragma unroll
  for (int k = 0; k < K_; ++k) {
    float acc = 0.0f;
#pragma unroll
    for (int d = 0; d < D_; ++d) {
      float df = xr[d] - sM[k * D_ + d];
      acc += df * df;
    }
    logit[k] = -acc * sc;
  }
  float mx = logit[0];
#pragma unroll
  for (int k = 1; k < K_; ++k) mx = fmaxf(mx, logit[k]);
  float e[K_]; float sum = 0.0f;
#pragma unroll
  for (int k = 0; k < K_; ++k) { e[k] = __expf(logit[k] - mx); sum += e[k]; }
  float inv = 1.0f / sum;
  float4 r = make_float4(e[0] * inv, e[1] * inv, e[2] * inv, e[3] * inv);
  *reinterpret_cast<float4*>(out + ((size_t)b * N + n) * K_) = r;
}

// ---------------------------------------------------------------------------
extern "C" void kernel_launch(void* const* d_in, const int* in_sizes, int n_in,
                              void* d_out, int out_size, void* d_ws, size_t ws_size,
                              hipStream_t stream) {
  const float* data = (const float*)d_in[0];
  const int* init_idx = (const int*)d_in[1];
  const float* scale = (const float*)d_in[2];
  float* out = (float*)d_out;

  const int N = in_sizes[0] / (B_ * D_);   // 131072 for the reference shapes

  float* wsMeans = (float*)d_ws;           // B*K*D = 320 floats
  float* wsPart = wsMeans + B_ * K_ * D_;  // B * nBlocks * 84 floats

  // clamp block count to available workspace (deterministic for fixed ws_size)
  int nBlocks = MAXBLOCKS;
  {
    size_t availF = ws_size / sizeof(float);
    size_t partF = (availF > (size_t)(B_ * K_ * D_)) ? availF - B_ * K_ * D_ : 0;
    int cap = (int)(partF / ((size_t)B_ * PSZ));
    if (cap < 1) cap = 1;
    if (nBlocks > cap) nBlocks = cap;
  }

  kinit<<<1, 320, 0, stream>>>(data, init_idx, wsMeans, N);

  const size_t meansOutOff = (size_t)B_ * N * K_;
  for (int it = 0; it < 5; ++it) {
    kaccum<<<dim3(nBlocks, B_), 256, 0, stream>>>(data, wsMeans, scale, wsPart, N);
    kreduce<<<B_, 128, 0, stream>>>(wsPart, wsMeans,
                                    (it == 4) ? (out + meansOutOff) : nullptr,
                                    nBlocks);
  }

  kfinal<<<dim3((N + 255) / 256, B_), 256, 0, stream>>>(data, wsMeans, scale, out, N);
}